// MipRayMarcher2_66829691126261
// MI455X (gfx1250) — compile-verified
//
#include <hip/hip_runtime.h>
#include <stdint.h>

#define BDIM 256   // 8 wave32 per block
#define RPB  8     // rays per block (one wave per ray)
#define NS   48    // samples per ray
#define NM   47    // mid samples per ray

// ws cells (ordered-uint encoding), spread across distinct cachelines so the
// per-block atomics to the 4 counters proceed in parallel in L2.
#define WS_DMIN 0
#define WS_DMAX 16
#define WS_RMIN 32
#define WS_RMAX 48

// ---------- ordered-uint encoding for float min/max atomics ----------
__device__ __forceinline__ unsigned f2ord(float f) {
  unsigned u = __float_as_uint(f);
  return (u & 0x80000000u) ? ~u : (u | 0x80000000u);
}
__device__ __forceinline__ float ord2f(unsigned u) {
  return (u & 0x80000000u) ? __uint_as_float(u ^ 0x80000000u) : __uint_as_float(~u);
}

__global__ void init_minmax_kernel(unsigned* __restrict__ ws) {
  ws[WS_DMIN] = 0xFFFFFFFFu;
  ws[WS_DMAX] = 0x00000000u;
  ws[WS_RMIN] = 0xFFFFFFFFu;
  ws[WS_RMAX] = 0x00000000u;
}

// ---------- async global -> LDS staging (gfx1250 ASYNCcnt path) ----------
// Each enabled lane copies 16B from global to its LDS byte address.
// LDS byte offset = low 32 bits of the flat shared-memory address (ISA 10.2).
__device__ __forceinline__ void stage_async(float* dst, const float* __restrict__ src,
                                            int nfloats, int tid) {
  for (int i = tid * 4; i < nfloats; i += BDIM * 4) {
    unsigned lds         = (unsigned)(uintptr_t)(dst + i);
    unsigned long long g = (unsigned long long)(uintptr_t)(src + i);
    asm volatile("global_load_async_to_lds_b128 %0, %1, off"
                 :: "v"(lds), "v"(g)
                 : "memory");
  }
}

__global__ __launch_bounds__(BDIM) void raymarch_kernel(
    const float* __restrict__ colors,   // (BR, NS, 3)
    const float* __restrict__ sdfs,     // (BR, NS)
    const float* __restrict__ depths,   // (BR, NS)
    const float* __restrict__ normals,  // (BR, NS, 3)
    const float* __restrict__ rdirs,    // (BR, 3)
    const float* __restrict__ rnorm,    // (BR, NS, 3)
    const float* __restrict__ inv_std_p,
    unsigned* __restrict__ ws,
    float* __restrict__ out_rgb,        // (BR, 3)
    float* __restrict__ out_depth,      // (BR)  unclipped (clip pass follows)
    float* __restrict__ out_w,          // (BR, NM)
    float* __restrict__ out_norm)       // (BR, 3) unclipped
{
  __shared__ __align__(16) float s_col[RPB * NS * 3];
  __shared__ __align__(16) float s_nrm[RPB * NS * 3];
  __shared__ __align__(16) float s_rnm[RPB * NS * 3];
  __shared__ __align__(16) float s_sdf[RPB * NS];
  __shared__ __align__(16) float s_dep[RPB * NS];
  __shared__ __align__(16) float s_dir[RPB * 3];
  __shared__ unsigned sh_red[4][RPB];

  const int  tid  = threadIdx.x;
  const long ray0 = (long)blockIdx.x * RPB;

  stage_async(s_col, colors + ray0 * NS * 3, RPB * NS * 3, tid);
  stage_async(s_nrm, normals + ray0 * NS * 3, RPB * NS * 3, tid);
  stage_async(s_rnm, rnorm  + ray0 * NS * 3, RPB * NS * 3, tid);
  stage_async(s_sdf, sdfs   + ray0 * NS,     RPB * NS,     tid);
  stage_async(s_dep, depths + ray0 * NS,     RPB * NS,     tid);
  stage_async(s_dir, rdirs  + ray0 * 3,      RPB * 3,      tid);
  asm volatile("s_wait_asynccnt 0x0" ::: "memory");
  __syncthreads();

  const int lane = tid & 31;
  const int wv   = tid >> 5;
  const long br  = ray0 + wv;

  // ---- block-local min/max of staged depths / real_normals (for clip pass)
  {
    unsigned dmn = 0xFFFFFFFFu, dmx = 0u, rmn = 0xFFFFFFFFu, rmx = 0u;
    for (int i = tid; i < RPB * NS; i += BDIM) {
      unsigned o = f2ord(s_dep[i]);
      dmn = min(dmn, o); dmx = max(dmx, o);
    }
    for (int i = tid; i < RPB * NS * 3; i += BDIM) {
      unsigned o = f2ord(s_rnm[i]);
      rmn = min(rmn, o); rmx = max(rmx, o);
    }
    #pragma unroll
    for (int off = 16; off >= 1; off >>= 1) {
      dmn = min(dmn, (unsigned)__shfl_xor((int)dmn, off, 32));
      dmx = max(dmx, (unsigned)__shfl_xor((int)dmx, off, 32));
      rmn = min(rmn, (unsigned)__shfl_xor((int)rmn, off, 32));
      rmx = max(rmx, (unsigned)__shfl_xor((int)rmx, off, 32));
    }
    if (lane == 0) {
      sh_red[0][wv] = dmn; sh_red[1][wv] = dmx;
      sh_red[2][wv] = rmn; sh_red[3][wv] = rmx;
    }
    __syncthreads();
    if (tid == 0) {
      unsigned a = sh_red[0][0], b = sh_red[1][0], c = sh_red[2][0], e = sh_red[3][0];
      #pragma unroll
      for (int i = 1; i < RPB; ++i) {
        a = min(a, sh_red[0][i]); b = max(b, sh_red[1][i]);
        c = min(c, sh_red[2][i]); e = max(e, sh_red[3][i]);
      }
      atomicMin(&ws[WS_DMIN], a); atomicMax(&ws[WS_DMAX], b);
      atomicMin(&ws[WS_RMIN], c); atomicMax(&ws[WS_RMAX], e);
    }
  }

  const float dx = s_dir[wv * 3 + 0];
  const float dy = s_dir[wv * 3 + 1];
  const float dz = s_dir[wv * 3 + 2];

  float inv_std = __expf(10.0f * inv_std_p[0]);
  inv_std = fminf(fmaxf(inv_std, 1e-6f), 1e6f);

  const float* dep = s_dep + wv * NS;
  const float* sdf = s_sdf + wv * NS;
  const float* col = s_col + wv * NS * 3;
  const float* nrm = s_nrm + wv * NS * 3;
  const float* rnm = s_rnm + wv * NS * 3;

  auto midcalc = [&](int m, float& a, float& c0, float& c1, float& c2,
                     float& dm, float& r0, float& r1, float& r2) {
    float d0 = dep[m], d1 = dep[m + 1];
    float delta = d1 - d0;
    dm = 0.5f * (d0 + d1);
    float sm = 0.5f * (sdf[m] + sdf[m + 1]);
    float n0 = 0.5f * (nrm[m * 3 + 0] + nrm[(m + 1) * 3 + 0]);
    float n1 = 0.5f * (nrm[m * 3 + 1] + nrm[(m + 1) * 3 + 1]);
    float n2 = 0.5f * (nrm[m * 3 + 2] + nrm[(m + 1) * 3 + 2]);
    c0 = 0.5f * (col[m * 3 + 0] + col[(m + 1) * 3 + 0]);
    c1 = 0.5f * (col[m * 3 + 1] + col[(m + 1) * 3 + 1]);
    c2 = 0.5f * (col[m * 3 + 2] + col[(m + 1) * 3 + 2]);
    r0 = 0.5f * (rnm[m * 3 + 0] + rnm[(m + 1) * 3 + 0]);
    r1 = 0.5f * (rnm[m * 3 + 1] + rnm[(m + 1) * 3 + 1]);
    r2 = 0.5f * (rnm[m * 3 + 2] + rnm[(m + 1) * 3 + 2]);
    float tc = dx * n0 + dy * n1 + dz * n2;
    float ic = fminf(tc, 0.0f);                 // -relu(-tc)
    float hid = 0.5f * ic * delta;
    float pc = 1.0f / (1.0f + __expf(-(sm - hid) * inv_std));
    float nc = 1.0f / (1.0f + __expf(-(sm + hid) * inv_std));
    a = fminf(fmaxf((pc - nc + 1e-5f) / (pc + 1e-5f), 0.0f), 1.0f);
  };

  // chunk 0: m = lane (0..31), chunk 1: m = lane + 32 (valid for lane < 15)
  float a0, c00, c01, c02, dm0, r00, r01, r02;
  midcalc(lane, a0, c00, c01, c02, dm0, r00, r01, r02);
  const bool v1 = (lane + 32) < NM;
  const int  m1 = v1 ? (lane + 32) : 0;
  float a1, c10, c11, c12, dm1, r10, r11, r12;
  midcalc(m1, a1, c10, c11, c12, dm1, r10, r11, r12);

  // wave32 inclusive prefix product of (1 - alpha + 1e-10)
  float incl0 = 1.0f - a0 + 1e-10f;
  #pragma unroll
  for (int d = 1; d < 32; d <<= 1) {
    float t = __shfl_up(incl0, d, 32);
    if (lane >= d) incl0 *= t;
  }
  float trans0 = __shfl_up(incl0, 1, 32);
  if (lane == 0) trans0 = 1.0f;
  float total0 = __shfl(incl0, 31, 32);

  float incl1 = v1 ? (1.0f - a1 + 1e-10f) : 1.0f;
  #pragma unroll
  for (int d = 1; d < 32; d <<= 1) {
    float t = __shfl_up(incl1, d, 32);
    if (lane >= d) incl1 *= t;
  }
  float trans1 = __shfl_up(incl1, 1, 32);
  if (lane == 0) trans1 = 1.0f;
  trans1 *= total0;

  float w0 = a0 * trans0;
  float w1 = v1 ? (a1 * trans1) : 0.0f;
  out_w[br * NM + lane] = w0;
  if (v1) out_w[br * NM + 32 + lane] = w1;

  // weighted partial sums per lane, then butterfly reduction
  float vr0 = w0 * c00 + w1 * c10;
  float vr1 = w0 * c01 + w1 * c11;
  float vr2 = w0 * c02 + w1 * c12;
  float vd  = w0 * dm0 + w1 * dm1;
  float vw  = w0 + w1;
  float vn0 = w0 * r00 + w1 * r10;
  float vn1 = w0 * r01 + w1 * r11;
  float vn2 = w0 * r02 + w1 * r12;
  #pragma unroll
  for (int off = 16; off >= 1; off >>= 1) {
    vr0 += __shfl_xor(vr0, off, 32);
    vr1 += __shfl_xor(vr1, off, 32);
    vr2 += __shfl_xor(vr2, off, 32);
    vd  += __shfl_xor(vd,  off, 32);
    vw  += __shfl_xor(vw,  off, 32);
    vn0 += __shfl_xor(vn0, off, 32);
    vn1 += __shfl_xor(vn1, off, 32);
    vn2 += __shfl_xor(vn2, off, 32);
  }

  if (lane == 0) {
    out_rgb[br * 3 + 0] = vr0;
    out_rgb[br * 3 + 1] = vr1;
    out_rgb[br * 3 + 2] = vr2;
    out_depth[br] = vd / vw;          // may be NaN; clip pass fixes up
    out_norm[br * 3 + 0] = vn0 / vw;
    out_norm[br * 3 + 1] = vn1 / vw;
    out_norm[br * 3 + 2] = vn2 / vw;
  }
}

// nan_to_num(nan=inf) then clip against the global min/max gathered above.
__global__ __launch_bounds__(BDIM) void clip_kernel(
    float* __restrict__ out_depth, float* __restrict__ out_norm,
    const unsigned* __restrict__ ws, int nrays) {
  int i = blockIdx.x * blockDim.x + threadIdx.x;
  if (i >= nrays) return;
  const float dmin = ord2f(ws[WS_DMIN]), dmax = ord2f(ws[WS_DMAX]);
  const float rmin = ord2f(ws[WS_RMIN]), rmax = ord2f(ws[WS_RMAX]);
  float d = out_depth[i];
  d = (d != d) ? dmax : fminf(fmaxf(d, dmin), dmax);   // nan->inf->clip = dmax
  out_depth[i] = d;
  #pragma unroll
  for (int c = 0; c < 3; ++c) {
    float n = out_norm[i * 3 + c];
    n = (n != n) ? rmax : fminf(fmaxf(n, rmin), rmax);
    out_norm[i * 3 + c] = n;
  }
}

extern "C" void kernel_launch(void* const* d_in, const int* in_sizes, int n_in,
                              void* d_out, int out_size, void* d_ws, size_t ws_size,
                              hipStream_t stream) {
  const float* colors  = (const float*)d_in[0];
  const float* sdfs    = (const float*)d_in[1];
  const float* depths  = (const float*)d_in[2];
  const float* normals = (const float*)d_in[3];
  const float* rdirs   = (const float*)d_in[4];
  const float* rnorm   = (const float*)d_in[5];
  const float* invstd  = (const float*)d_in[6];
  unsigned* ws = (unsigned*)d_ws;

  const long nrays = in_sizes[4] / 3;      // B*R

  float* out     = (float*)d_out;
  float* out_rgb = out;
  float* out_dep = out + nrays * 3;
  float* out_w   = out + nrays * 4;
  float* out_nrm = out + nrays * 4 + nrays * NM;

  hipLaunchKernelGGL(init_minmax_kernel, dim3(1), dim3(1), 0, stream, ws);
  hipLaunchKernelGGL(raymarch_kernel, dim3((unsigned)(nrays / RPB)), dim3(BDIM), 0, stream,
                     colors, sdfs, depths, normals, rdirs, rnorm, invstd, ws,
                     out_rgb, out_dep, out_w, out_nrm);
  hipLaunchKernelGGL(clip_kernel, dim3((unsigned)((nrays + BDIM - 1) / BDIM)), dim3(BDIM),
                     0, stream, out_dep, out_nrm, ws, (int)nrays);
}